// PeerRegularizationLayerAtt_64407329571369
// MI455X (gfx1250) — compile-verified
//
#include <hip/hip_runtime.h>
#include <hip/hip_bf16.h>
#include <math.h>

// ---------------------------------------------------------------------------
// PeerRegularizationLayerAtt for MI455X (gfx1250, wave32, WMMA)
//
// Heavy op: S = xi @ xp^T (4096 x 16384 x 256, fp32) via
// V_WMMA_F32_16X16X4_F32. Everything downstream of the KNN mask is 5-sparse
// per row (exp(-1e9) == 0 in fp32), so att@sp collapses to a 5-row gather.
//
// Main kernel shape: 32 output rows per block, each wave computes two
// stacked 16x16 tiles (rows 0-15 / 16-31) sharing one B fragment
// -> 1 global b32 load per wmma, A fragments from LDS.
// ---------------------------------------------------------------------------

typedef float v2f __attribute__((ext_vector_type(2)));
typedef float v8f __attribute__((ext_vector_type(8)));

#define C_IN   256
#define H1D    128
#define H2D    64
#define PIX    4096          // 64*64
#define N_ROWS 4096          // inp pixels (B=1)
#define M_ROWS 16384         // peer pixels (B=4)
#define NCHUNK 4             // one chunk per peer batch
#define NWAVES 8
#define BROWS  32            // rows per block
#define EPSF   1e-5f
#define BIGF   3.0e38f

// --------------------------- WMMA wrapper ----------------------------------
// D(16x16,f32) += A(16x4,f32) * B(4x16,f32), fp32 accumulate (RNE).
__device__ __forceinline__ v8f wmma_f32_16x16x4(v2f a, v2f b, v8f c) {
  return __builtin_amdgcn_wmma_f32_16x16x4_f32(
      /*neg_a=*/false, a, /*neg_b=*/false, b,
      /*c_mod=*/(short)0, c, /*reuse_a=*/false, /*reuse_b=*/false);
}

// --------------------------- top-5 (ascending) -----------------------------
struct Top5 { float v0, v1, v2, v3, v4; int i0, i1, i2, i3, i4; };

__device__ __forceinline__ void top5_init(Top5& t) {
  t.v0 = t.v1 = t.v2 = t.v3 = t.v4 = BIGF;
  t.i0 = t.i1 = t.i2 = t.i3 = t.i4 = 0;
}

// Branchless sorted insert, caller guarantees v < t.v4.
__device__ __forceinline__ void top5_shift_ins(Top5& t, float v, int j) {
  const bool c3 = v < t.v3, c2 = v < t.v2, c1 = v < t.v1, c0 = v < t.v0;
  const float nv4 = c3 ? t.v3 : v;              const int ni4 = c3 ? t.i3 : j;
  const float nv3 = c2 ? t.v2 : (c3 ? v : t.v3); const int ni3 = c2 ? t.i2 : (c3 ? j : t.i3);
  const float nv2 = c1 ? t.v1 : (c2 ? v : t.v2); const int ni2 = c1 ? t.i1 : (c2 ? j : t.i2);
  const float nv1 = c0 ? t.v0 : (c1 ? v : t.v1); const int ni1 = c0 ? t.i0 : (c1 ? j : t.i1);
  const float nv0 = c0 ? v : t.v0;               const int ni0 = c0 ? j : t.i0;
  t.v4 = nv4; t.i4 = ni4; t.v3 = nv3; t.i3 = ni3; t.v2 = nv2; t.i2 = ni2;
  t.v1 = nv1; t.i1 = ni1; t.v0 = nv0; t.i0 = ni0;
}

__device__ __forceinline__ void top5_ins(Top5& t, float v, int j) {
  if (v < t.v4) top5_shift_ins(t, v, j);
}

// --------------------------- tiny helpers ----------------------------------
__device__ __forceinline__ float softplus_stable(float x) {
  return fmaxf(x, 0.0f) + log1pf(expf(-fabsf(x)));
}

// pixel-major fetch base: x[r, :] for [B,C,64,64] tensors, r = global pixel
__device__ __forceinline__ const float* pix_base(const float* x, int r) {
  return x + (r >> 12) * (C_IN * PIX) + (r & (PIX - 1));
}

// --------------------------- utility kernels -------------------------------
__global__ __launch_bounds__(256) void k_copy4(const float* __restrict__ src,
                                               float* __restrict__ dst, int n4) {
  int i = blockIdx.x * 256 + threadIdx.x;
  if (i < n4) reinterpret_cast<float4*>(dst)[i] =
      reinterpret_cast<const float4*>(src)[i];
}

__global__ __launch_bounds__(256) void k_colnorm(const float* __restrict__ xp,
                                                 float* __restrict__ nj) {
  int j = blockIdx.x * 256 + threadIdx.x;
  const float* base = pix_base(xp, j);
  float s = 0.0f;
#pragma unroll 4
  for (int c = 0; c < C_IN; ++c) { float v = base[c * PIX]; s = fmaf(v, v, s); }
  nj[j] = s;
}

// H[o][r] = sum_c X[r,c] * W[o,c] + b[o]   (X in pixel-major BCHW layout)
__global__ __launch_bounds__(256) void k_gemm_in(const float* __restrict__ X,
                                                 const float* __restrict__ W,
                                                 const float* __restrict__ b,
                                                 float* __restrict__ H, int R) {
  int r = blockIdx.x * 256 + threadIdx.x;
  int o = blockIdx.y;
  const float* w = W + o * C_IN;
  const float* base = pix_base(X, r);
  float acc = b[o];
#pragma unroll 4
  for (int c = 0; c < C_IN; ++c) acc = fmaf(base[c * PIX], w[c], acc);
  H[o * R + r] = acc;
}

// H[o][r] = sum_c U[c][r] * W[o,c] + b[o]
__global__ __launch_bounds__(256) void k_gemm_mid(const float* __restrict__ U,
                                                  const float* __restrict__ W,
                                                  const float* __restrict__ b,
                                                  float* __restrict__ H,
                                                  int R, int K) {
  int r = blockIdx.x * 256 + threadIdx.x;
  int o = blockIdx.y;
  const float* w = W + o * K;
  float acc = b[o];
#pragma unroll 4
  for (int c = 0; c < K; ++c) acc = fmaf(U[c * R + r], w[c], acc);
  H[o * R + r] = acc;
}

// biased batch mean/var per column (two-pass, matches jnp.mean/var numerics)
__global__ __launch_bounds__(256) void k_meanvar(const float* __restrict__ H,
                                                 float* __restrict__ mean,
                                                 float* __restrict__ var, int R) {
  __shared__ float red[256];
  int col = blockIdx.x;
  const float* base = H + col * R;
  float s = 0.0f;
  for (int r = threadIdx.x; r < R; r += 256) s += base[r];
  red[threadIdx.x] = s;
  __syncthreads();
  for (int st = 128; st > 0; st >>= 1) {
    if (threadIdx.x < st) red[threadIdx.x] += red[threadIdx.x + st];
    __syncthreads();
  }
  float m = red[0] / (float)R;
  __syncthreads();
  float s2 = 0.0f;
  for (int r = threadIdx.x; r < R; r += 256) { float d = base[r] - m; s2 = fmaf(d, d, s2); }
  red[threadIdx.x] = s2;
  __syncthreads();
  for (int st = 128; st > 0; st >>= 1) {
    if (threadIdx.x < st) red[threadIdx.x] += red[threadIdx.x + st];
    __syncthreads();
  }
  if (threadIdx.x == 0) { mean[col] = m; var[col] = red[0] / (float)R; }
}

__global__ __launch_bounds__(256) void k_bnrelu(float* __restrict__ H,
                                                const float* __restrict__ mean,
                                                const float* __restrict__ var,
                                                const float* __restrict__ g,
                                                const float* __restrict__ be, int R) {
  int r = blockIdx.x * 256 + threadIdx.x;
  int col = blockIdx.y;
  float rs = rsqrtf(var[col] + EPSF);
  float x = H[col * R + r];
  x = (x - mean[col]) * rs * g[col] + be[col];
  H[col * R + r] = fmaxf(x, 0.0f);
}

__global__ __launch_bounds__(256) void k_dot(const float* __restrict__ T,
                                             const float* __restrict__ cw,
                                             const float* __restrict__ cb,
                                             float* __restrict__ a, int R, int K) {
  int r = blockIdx.x * 256 + threadIdx.x;
  float acc = cb[0];
#pragma unroll 4
  for (int o = 0; o < K; ++o) acc = fmaf(T[o * R + r], cw[o], acc);
  a[r] = acc;
}

// --------------------------- main WMMA kernel ------------------------------
// score s(i,j) = nj[j] - 2 * <xi_i, xp_j>  (same ranking as euclidean dist).
// grid = (NCHUNK, N_ROWS/32); block = 256 (8 waves). Each wave computes two
// stacked 16x16 tiles (block rows 0-15 and 16-31) that SHARE the B fragment:
// per k-step = 2 ds_load_2addr (A0,A1) + 2 global b32 (B) + 2 wmma.
// Lane L then maintains the running top-5 for block row L through a per-wave
// LDS score tile (pitch 19 -> conflict-free writes and row scans).
__global__ __launch_bounds__(256, 1)
void k_dist_topk(const float* __restrict__ xiT,   // inp_cont  [256][4096]
                 const float* __restrict__ xpT,   // peers_cont[4][256][4096]
                 const float* __restrict__ nj,    // [16384]
                 float* __restrict__ pv,          // [4096][NCHUNK][5]
                 int* __restrict__ pi) {          // [4096][NCHUNK][5]
  __shared__ float Alds[C_IN][BROWS];        // -2 * xi^T stripe, [k][m]
  __shared__ float tile[NWAVES][BROWS][19];  // per-wave score tile
  __shared__ float mrgV[NWAVES][BROWS][5];
  __shared__ int   mrgI[NWAVES][BROWS][5];

  const int i0    = blockIdx.y * BROWS;
  const int chunk = blockIdx.x;                 // == peer batch index
  const float* xpB = xpT + chunk * (C_IN * PIX);
  const int tid  = threadIdx.x;
  const int wave = tid >> 5;
  const int lane = tid & 31;
  const int half = lane >> 4;
  const int qc   = lane & 15;

  // Stage A stripe: Alds[k][m] = -2 * xi[i0+m, k]
  for (int idx = tid; idx < C_IN * BROWS; idx += 256) {
    int k = idx >> 5, m = idx & 31;
    Alds[k][m] = -2.0f * xiT[k * PIX + i0 + m];
  }
  __syncthreads();

  Top5 t5; top5_init(t5);   // lane L owns block row L

  for (int jt = wave * 16; jt < PIX; jt += NWAVES * 16) {
    v8f acc0 = {0.f, 0.f, 0.f, 0.f, 0.f, 0.f, 0.f, 0.f};
    v8f acc1 = {0.f, 0.f, 0.f, 0.f, 0.f, 0.f, 0.f, 0.f};
    const float* bp = xpB + jt + qc;

#pragma unroll 4
    for (int k0 = 0; k0 < C_IN; k0 += 4) {
      const int kk = k0 + 2 * half;
      v2f a0; a0.x = Alds[kk][qc];       a0.y = Alds[kk + 1][qc];
      v2f a1; a1.x = Alds[kk][16 + qc];  a1.y = Alds[kk + 1][16 + qc];
      v2f bv; bv.x = bp[kk * PIX];       bv.y = bp[(kk + 1) * PIX];
      acc0 = wmma_f32_16x16x4(a0, bv, acc0);  // rows  0-15
      acc1 = wmma_f32_16x16x4(a1, bv, acc1);  // rows 16-31
    }

    const float njv = nj[chunk * PIX + jt + qc];

    // scores -> LDS (ISA C/D layout: D[m][n], m = r + 8*half, n = qc)
#pragma unroll
    for (int r = 0; r < 8; ++r) tile[wave][r + 8 * half][qc]      = acc0[r] + njv;
#pragma unroll
    for (int r = 0; r < 8; ++r) tile[wave][16 + r + 8 * half][qc] = acc1[r] + njv;

    // same-wave LDS ops are in-order: the row scan sees all lanes' writes.
    const int jb = chunk * PIX + jt;
#pragma unroll
    for (int t = 0; t < 16; ++t) {
      float v = tile[wave][lane][t];
      if (v < t5.v4) top5_shift_ins(t5, v, jb + t);   // rare after warm-up
    }
  }

  mrgV[wave][lane][0] = t5.v0; mrgI[wave][lane][0] = t5.i0;
  mrgV[wave][lane][1] = t5.v1; mrgI[wave][lane][1] = t5.i1;
  mrgV[wave][lane][2] = t5.v2; mrgI[wave][lane][2] = t5.i2;
  mrgV[wave][lane][3] = t5.v3; mrgI[wave][lane][3] = t5.i3;
  mrgV[wave][lane][4] = t5.v4; mrgI[wave][lane][4] = t5.i4;
  __syncthreads();

  if (tid < BROWS) {
    const int row = tid;
    Top5 m; top5_init(m);
    for (int w = 0; w < NWAVES; ++w)
#pragma unroll
      for (int e = 0; e < 5; ++e) top5_ins(m, mrgV[w][row][e], mrgI[w][row][e]);
    const int bp = ((i0 + row) * NCHUNK + chunk) * 5;
    pv[bp + 0] = m.v0; pi[bp + 0] = m.i0;
    pv[bp + 1] = m.v1; pi[bp + 1] = m.i1;
    pv[bp + 2] = m.v2; pi[bp + 2] = m.i2;
    pv[bp + 3] = m.v3; pi[bp + 3] = m.i3;
    pv[bp + 4] = m.v4; pi[bp + 4] = m.i4;
  }
}

// --------------------------- finalize --------------------------------------
// Merge chunk top-5s, softmax over the 5 survivors (masked terms are exactly
// 0 in fp32), weighted gather of peers_style rows into the concat output.
__global__ __launch_bounds__(256)
void k_finalize(const float* __restrict__ pv, const int* __restrict__ pi,
                const float* __restrict__ a1, const float* __restrict__ a2,
                const float* __restrict__ ps,   // peers_style
                float* __restrict__ outF) {     // d_out + 256*4096
  __shared__ float wgt[64][5];
  __shared__ int   jj[64][5];
  const int r0  = blockIdx.x * 64;
  const int tid = threadIdx.x;

  if (tid < 64) {
    const int row = r0 + tid;
    Top5 m; top5_init(m);
    for (int c = 0; c < NCHUNK; ++c)
#pragma unroll
      for (int e = 0; e < 5; ++e) {
        const int bp = (row * NCHUNK + c) * 5 + e;
        top5_ins(m, pv[bp], pi[bp]);
      }
    const float av = a1[row];
    float s0 = softplus_stable(av + a2[m.i0]);
    float s1 = softplus_stable(av + a2[m.i1]);
    float s2 = softplus_stable(av + a2[m.i2]);
    float s3 = softplus_stable(av + a2[m.i3]);
    float s4 = softplus_stable(av + a2[m.i4]);
    float mx = fmaxf(fmaxf(fmaxf(s0, s1), fmaxf(s2, s3)), s4);
    float w0 = expf(s0 - mx), w1 = expf(s1 - mx), w2 = expf(s2 - mx);
    float w3 = expf(s3 - mx), w4 = expf(s4 - mx);
    float inv = 1.0f / (w0 + w1 + w2 + w3 + w4);
    wgt[tid][0] = w0 * inv; jj[tid][0] = m.i0;
    wgt[tid][1] = w1 * inv; jj[tid][1] = m.i1;
    wgt[tid][2] = w2 * inv; jj[tid][2] = m.i2;
    wgt[tid][3] = w3 * inv; jj[tid][3] = m.i3;
    wgt[tid][4] = w4 * inv; jj[tid][4] = m.i4;
  }
  __syncthreads();

  const int csub = tid >> 6;     // 0..3
  const int p    = tid & 63;     // local row
  const float w0 = wgt[p][0], w1 = wgt[p][1], w2 = wgt[p][2],
              w3 = wgt[p][3], w4 = wgt[p][4];
  const int j0 = jj[p][0], j1 = jj[p][1], j2 = jj[p][2],
            j3 = jj[p][3], j4 = jj[p][4];
  const int b0 = (j0 >> 12) * (C_IN * PIX) + (j0 & (PIX - 1));
  const int b1 = (j1 >> 12) * (C_IN * PIX) + (j1 & (PIX - 1));
  const int b2 = (j2 >> 12) * (C_IN * PIX) + (j2 & (PIX - 1));
  const int b3 = (j3 >> 12) * (C_IN * PIX) + (j3 & (PIX - 1));
  const int b4 = (j4 >> 12) * (C_IN * PIX) + (j4 & (PIX - 1));

  for (int c = csub; c < C_IN; c += 4) {
    const int co = c * PIX;
    float acc = w0 * ps[b0 + co];
    acc = fmaf(w1, ps[b1 + co], acc);
    acc = fmaf(w2, ps[b2 + co], acc);
    acc = fmaf(w3, ps[b3 + co], acc);
    acc = fmaf(w4, ps[b4 + co], acc);
    outF[co + r0 + p] = acc;   // fmap[0, c, pix] with pix == row
  }
}

// --------------------------- host ------------------------------------------
extern "C" void kernel_launch(void* const* d_in, const int* in_sizes, int n_in,
                              void* d_out, int out_size, void* d_ws, size_t ws_size,
                              hipStream_t stream) {
  (void)in_sizes; (void)n_in; (void)out_size; (void)ws_size;
  const float* inp_cont    = (const float*)d_in[0];
  /* d_in[1] = inp_style: shape-only in reference */
  const float* peers_cont  = (const float*)d_in[2];
  const float* peers_style = (const float*)d_in[3];
  const float* W1  = (const float*)d_in[4];
  const float* b1  = (const float*)d_in[5];
  const float* g1  = (const float*)d_in[6];
  const float* be1 = (const float*)d_in[7];
  const float* W2  = (const float*)d_in[8];
  const float* b2  = (const float*)d_in[9];
  const float* g2  = (const float*)d_in[10];
  const float* be2 = (const float*)d_in[11];
  const float* cw1 = (const float*)d_in[12];
  const float* cb1 = (const float*)d_in[13];
  const float* cw2 = (const float*)d_in[14];
  const float* cb2 = (const float*)d_in[15];
  float* out = (float*)d_out;

  float* ws = (float*)d_ws;
  size_t off = 0;
  float* h1 = ws + off; off += (size_t)M_ROWS * H1D;           // 8 MB
  float* h2 = ws + off; off += (size_t)M_ROWS * H2D;           // 4 MB
  float* nj = ws + off; off += M_ROWS;
  float* a1 = ws + off; off += N_ROWS;
  float* a2 = ws + off; off += M_ROWS;
  float* m1 = ws + off; off += H1D;
  float* v1 = ws + off; off += H1D;
  float* m2 = ws + off; off += H2D;
  float* v2 = ws + off; off += H2D;
  float* pv = ws + off; off += (size_t)N_ROWS * NCHUNK * 5;
  int*   pi = (int*)(ws + off);

  // concat part 1: copy inp_cont
  k_copy4<<<(C_IN * PIX / 4 + 255) / 256, 256, 0, stream>>>(inp_cont, out,
                                                            C_IN * PIX / 4);
  // peer squared norms
  k_colnorm<<<M_ROWS / 256, 256, 0, stream>>>(peers_cont, nj);

  auto transform = [&](const float* X, int R, const float* cw, const float* cb,
                       float* aout) {
    k_gemm_in<<<dim3(R / 256, H1D), 256, 0, stream>>>(X, W1, b1, h1, R);
    k_meanvar<<<H1D, 256, 0, stream>>>(h1, m1, v1, R);
    k_bnrelu<<<dim3(R / 256, H1D), 256, 0, stream>>>(h1, m1, v1, g1, be1, R);
    k_gemm_mid<<<dim3(R / 256, H2D), 256, 0, stream>>>(h1, W2, b2, h2, R, H1D);
    k_meanvar<<<H2D, 256, 0, stream>>>(h2, m2, v2, R);
    k_bnrelu<<<dim3(R / 256, H2D), 256, 0, stream>>>(h2, m2, v2, g2, be2, R);
    k_dot<<<R / 256, 256, 0, stream>>>(h2, cw, cb, aout, R, H2D);
  };
  transform(inp_cont,   N_ROWS, cw1, cb1, a1);   // separate batch stats,
  transform(peers_cont, M_ROWS, cw2, cb2, a2);   // matching the two _transf calls

  // fused distance GEMM + per-row top-5 (WMMA fp32)
  k_dist_topk<<<dim3(NCHUNK, N_ROWS / BROWS), 256, 0, stream>>>(
      inp_cont, peers_cont, nj, pv, pi);

  // merge + 5-way softmax + weighted gather into concat part 2
  k_finalize<<<N_ROWS / 64, 256, 0, stream>>>(pv, pi, a1, a2, peers_style,
                                              out + C_IN * PIX);
}